// TransformerBlockQuantum_5368709120608
// MI455X (gfx1250) — compile-verified
//
#include <hip/hip_runtime.h>

// ---------------------------------------------------------------------------
// Quantum transformer block, MI455X (gfx1250, wave32).
//
// Closed form of qlayer (product state + CNOT ring = prefix products of cos):
//   phi_q = (sum of encoder angles mapped to wire q) + theta_q
//   out[w] = prod_{q=0..w} cos(phi_q)   (w = 1..7)
//   out[0] = prod_{q=1..7} cos(phi_q)
//
// Pipeline:
//   K1: hbuf[t][0..7] = qlayer_attn(x[t])                  (elementwise)
//   K2: attention per (b, head): scores via V_WMMA_F32_16X16X4_F32 (dk = 4),
//       K/V staged to LDS with GLOBAL_LOAD_ASYNC_TO_LDS_B128 (ASYNCcnt),
//       softmax via raw v_exp_f32 (0.5*log2e folded into A; |score'| <= 2.89
//       so neither a running max nor the denormal-safe exp2 expansion is
//       needed), P*V accumulated per-lane, one cross-lane xor-shuffle
//       reduction at the end.
//   K3: per-token tail: qlayer_attn(ctx) + residual + LN1 + 8x8 GEMV +
//       qlayer_ffn + 8x8 GEMV + residual + LN2.
// ---------------------------------------------------------------------------

typedef float v2f __attribute__((ext_vector_type(2)));
typedef float v8f __attribute__((ext_vector_type(8)));

#define B_    128
#define T_    512
#define E_    8
#define NTOK  (B_ * T_)   // 65536

#define LOG2E 1.44269504088896340736f

// ------------------------- qlayer closed forms -----------------------------

__device__ __forceinline__ void qlayer_from_cos(const float c[8], float o[8]) {
    // o[w] = c0*...*cw for w>=1 ; o[0] = c1*...*c7
    float run = c[0];
#pragma unroll
    for (int w = 1; w < 8; ++w) { run *= c[w]; o[w] = run; }
    float s = c[7];
#pragma unroll
    for (int q = 6; q >= 1; --q) s *= c[q];
    o[0] = s;
}

__device__ __forceinline__ void qlayer_attn8(const float in[8], const float th[8], float o[8]) {
    float c[8];
#pragma unroll
    for (int q = 0; q < 8; ++q) c[q] = __cosf(in[q] + th[q]);
    qlayer_from_cos(c, o);
}

__device__ __forceinline__ void qlayer_ffn8(const float in[8], const float th[8], float o[8]) {
    // FFN_WIRES = [0,1,2,2,4,5,6,7]: wire2 gets in[2]+in[3], wire3 only theta.
    float c[8];
    c[0] = __cosf(in[0] + th[0]);
    c[1] = __cosf(in[1] + th[1]);
    c[2] = __cosf(in[2] + in[3] + th[2]);
    c[3] = __cosf(th[3]);
#pragma unroll
    for (int q = 4; q < 8; ++q) c[q] = __cosf(in[q] + th[q]);
    qlayer_from_cos(c, o);
}

__device__ __forceinline__ void layernorm8(const float y[8], const float* w,
                                           const float* b, float o[8]) {
    float m = 0.f;
#pragma unroll
    for (int i = 0; i < 8; ++i) m += y[i];
    m *= 0.125f;
    float v = 0.f;
#pragma unroll
    for (int i = 0; i < 8; ++i) { float d = y[i] - m; v += d * d; }
    v *= 0.125f;
    float r = __frsqrt_rn(v + 1e-5f);
#pragma unroll
    for (int i = 0; i < 8; ++i) o[i] = (y[i] - m) * r * w[i] + b[i];
}

// ------------------------- K1: qkv = qlayer(x) -----------------------------

__global__ __launch_bounds__(256) void k_qlayer_attn(const float* __restrict__ x,
                                                     const float* __restrict__ th,
                                                     float* __restrict__ hbuf) {
    int t = blockIdx.x * 256 + threadIdx.x;
    const float4* xr = (const float4*)(x + (size_t)t * 8);
    float4 a = xr[0], bb = xr[1];
    float in[8] = {a.x, a.y, a.z, a.w, bb.x, bb.y, bb.z, bb.w};
    float thl[8];
#pragma unroll
    for (int q = 0; q < 8; ++q) thl[q] = th[q];
    float o[8];
    qlayer_attn8(in, thl, o);
    float4* hr = (float4*)(hbuf + (size_t)t * 8);
    hr[0] = make_float4(o[0], o[1], o[2], o[3]);
    hr[1] = make_float4(o[4], o[5], o[6], o[7]);
}

// ------------------------- K2: attention (WMMA f32 16x16x4) ----------------
// grid: B * H * 4 blocks; block = 256 threads = 8 waves; each wave = one
// 16-query tile; 4 blocks cover the 32 query tiles of one (b, head).

__global__ __launch_bounds__(256) void k_attention(const float* __restrict__ hbuf,
                                                   float* __restrict__ ctx) {
    __shared__ float ks[T_ * 4];   // this head's rows: 512 x 4 f32 = 8 KB

    int bx = blockIdx.x;
    int qg = bx & 3;           // query-tile group 0..3
    int bh = bx >> 2;
    int h  = bh & 1;
    int b  = bh >> 1;

    const float* hb = hbuf + ((size_t)b * T_) * 8 + h * 4;

    // Stage K/V (= Q source too) into LDS with the CDNA5 async global->LDS
    // DMA path (no VGPR round-trip, tracked by ASYNCcnt). Each lane moves
    // one 16B row slice per iteration: 512 rows x 16B = 8 KB.
    for (int r = threadIdx.x; r < T_; r += 256) {
        unsigned ldsoff = (unsigned)(uintptr_t)(ks + r * 4);   // LDS byte offset
        const float* gp = hb + (size_t)r * 8;
        asm volatile("global_load_async_to_lds_b128 %0, %1, off"
                     :: "v"(ldsoff), "v"(gp) : "memory");
    }
    asm volatile("s_wait_asynccnt 0x0" ::: "memory");
    __syncthreads();

    int wave = threadIdx.x >> 5;
    int lane = threadIdx.x & 31;
    int half = lane >> 4;      // 0: rows r, 1: rows r+8 (C-layout halves)
    int l16  = lane & 15;
    int qt   = qg * 8 + wave;  // 16-query tile index, 0..31

    // A fragment (16x4 f32): lane L holds A[M=L%16][K = v + 2*(L/16)].
    // Fold 1/sqrt(dk) = 0.5 AND log2(e) into A once: then the softmax
    // numerator is exp2(S') = exp(S/2) exactly, and v_exp_f32 is native 2^x.
    int qrow = qt * 16 + l16;
    v2f afrag;
    afrag.x = ks[qrow * 4 + 2 * half + 0] * (0.5f * LOG2E);
    afrag.y = ks[qrow * 4 + 2 * half + 1] * (0.5f * LOG2E);

    float acc[8][4];
    float rs[8];
#pragma unroll
    for (int r = 0; r < 8; ++r) {
        rs[r] = 0.f;
#pragma unroll
        for (int d = 0; d < 4; ++d) acc[r][d] = 0.f;
    }

    for (int kt = 0; kt < T_ / 16; ++kt) {
        int key = kt * 16 + l16;
        // B fragment (4x16 f32): lane L holds B[K = v + 2*(L/16)][N=L%16].
        v2f bfrag;
        bfrag.x = ks[key * 4 + 2 * half + 0];
        bfrag.y = ks[key * 4 + 2 * half + 1];

        v8f cz = {};
        v8f s = __builtin_amdgcn_wmma_f32_16x16x4_f32(
            /*neg_a=*/false, afrag, /*neg_b=*/false, bfrag,
            /*c_mod=*/(short)0, cz, /*reuse_a=*/false, /*reuse_b=*/false);

        // V row for this lane's key: 4 features.
        float4 vv = *(const float4*)(ks + key * 4);

        // Raw v_exp_f32 (2^x). |s| <= 2*log2e, far from the subnormal
        // cutoff, so the libm denorm-guard expansion is unnecessary.
#pragma unroll
        for (int r = 0; r < 8; ++r) {
            float p = __builtin_amdgcn_exp2f(s[r]);
            rs[r] += p;
            acc[r][0] += p * vv.x;
            acc[r][1] += p * vv.y;
            acc[r][2] += p * vv.z;
            acc[r][3] += p * vv.w;
        }
    }

    // Reduce over the 16 key-lanes of each half (xor masks stay in-half).
#pragma unroll
    for (int m = 1; m <= 8; m <<= 1) {
#pragma unroll
        for (int r = 0; r < 8; ++r) {
            rs[r] += __shfl_xor(rs[r], m, 32);
#pragma unroll
            for (int d = 0; d < 4; ++d)
                acc[r][d] += __shfl_xor(acc[r][d], m, 32);
        }
    }

    // Lane 0 writes rows r (half 0); lane 16 writes rows r+8 (half 1).
    if (l16 == 0) {
#pragma unroll
        for (int r = 0; r < 8; ++r) {
            int row = qt * 16 + r + 8 * half;
            float inv = 1.0f / rs[r];
            float4 o = make_float4(acc[r][0] * inv, acc[r][1] * inv,
                                   acc[r][2] * inv, acc[r][3] * inv);
            *(float4*)(ctx + ((size_t)b * T_ + row) * 8 + h * 4) = o;
        }
    }
}

// ------------------------- K3: fused per-token tail ------------------------

__global__ __launch_bounds__(256) void k_tail(const float* __restrict__ x,
                                              const float* __restrict__ ctxbuf,
                                              const float* __restrict__ th_attn,
                                              const float* __restrict__ th_ffn,
                                              const float* __restrict__ W_in,
                                              const float* __restrict__ b_in,
                                              const float* __restrict__ W_out,
                                              const float* __restrict__ b_out,
                                              const float* __restrict__ ln1w,
                                              const float* __restrict__ ln1b,
                                              const float* __restrict__ ln2w,
                                              const float* __restrict__ ln2b,
                                              float* __restrict__ out) {
    int t = blockIdx.x * 256 + threadIdx.x;

    float cx[8], xr[8];
    {
        const float4* cr = (const float4*)(ctxbuf + (size_t)t * 8);
        float4 a = cr[0], bb = cr[1];
        cx[0]=a.x; cx[1]=a.y; cx[2]=a.z; cx[3]=a.w;
        cx[4]=bb.x; cx[5]=bb.y; cx[6]=bb.z; cx[7]=bb.w;
        const float4* xrp = (const float4*)(x + (size_t)t * 8);
        float4 c = xrp[0], d = xrp[1];
        xr[0]=c.x; xr[1]=c.y; xr[2]=c.z; xr[3]=c.w;
        xr[4]=d.x; xr[5]=d.y; xr[6]=d.z; xr[7]=d.w;
    }

    float tha[8], thf[8];
#pragma unroll
    for (int q = 0; q < 8; ++q) { tha[q] = th_attn[q]; thf[q] = th_ffn[q]; }

    // attn_out = qlayer(ctx), x1 = LN1(attn_out + x)
    float ao[8];
    qlayer_attn8(cx, tha, ao);
    float y[8];
#pragma unroll
    for (int i = 0; i < 8; ++i) y[i] = ao[i] + xr[i];
    float x1[8];
    layernorm8(y, ln1w, ln1b, x1);

    // hq = x1 @ W_in^T + b_in
    float hq[8];
#pragma unroll
    for (int i = 0; i < 8; ++i) {
        float s = b_in[i];
#pragma unroll
        for (int j = 0; j < 8; ++j) s = fmaf(W_in[i * 8 + j], x1[j], s);
        hq[i] = s;
    }

    // q = qlayer_ffn(hq); ff = q @ W_out^T + b_out
    float q8[8];
    qlayer_ffn8(hq, thf, q8);
    float z[8];
#pragma unroll
    for (int i = 0; i < 8; ++i) {
        float s = b_out[i];
#pragma unroll
        for (int j = 0; j < 8; ++j) s = fmaf(W_out[i * 8 + j], q8[j], s);
        z[i] = s + x1[i];   // residual
    }

    float o[8];
    layernorm8(z, ln2w, ln2b, o);

    float4* orow = (float4*)(out + (size_t)t * 8);
    orow[0] = make_float4(o[0], o[1], o[2], o[3]);
    orow[1] = make_float4(o[4], o[5], o[6], o[7]);
}

// ------------------------- launcher ----------------------------------------

extern "C" void kernel_launch(void* const* d_in, const int* in_sizes, int n_in,
                              void* d_out, int out_size, void* d_ws, size_t ws_size,
                              hipStream_t stream) {
    const float* x       = (const float*)d_in[0];
    const float* th_attn = (const float*)d_in[1];
    const float* th_ffn  = (const float*)d_in[2];
    const float* W_in    = (const float*)d_in[3];
    const float* b_in    = (const float*)d_in[4];
    const float* W_out   = (const float*)d_in[5];
    const float* b_out   = (const float*)d_in[6];
    const float* ln1w    = (const float*)d_in[7];
    const float* ln1b    = (const float*)d_in[8];
    const float* ln2w    = (const float*)d_in[9];
    const float* ln2b    = (const float*)d_in[10];

    float* hbuf = (float*)d_ws;                 // NTOK*8 f32 = 2 MB
    float* ctx  = hbuf + (size_t)NTOK * 8;      // NTOK*8 f32 = 2 MB

    k_qlayer_attn<<<NTOK / 256, 256, 0, stream>>>(x, th_attn, hbuf);
    k_attention<<<B_ * 2 * 4, 256, 0, stream>>>(hbuf, ctx);
    k_tail<<<NTOK / 256, 256, 0, stream>>>(x, ctx, th_attn, th_ffn,
                                           W_in, b_in, W_out, b_out,
                                           ln1w, ln1b, ln2w, ln2b,
                                           (float*)d_out);
}